// CRF_43525198578044
// MI455X (gfx1250) — compile-verified
//
#include <hip/hip_runtime.h>
#include <math.h>

// Problem constants (match reference: B=512, T=1024, C=48)
#define Bsz  512
#define Tlen 1024
#define Ccls 48
#define NT   3    // N tiles of 16 (3*16 = 48 classes)
#define KC   12   // K chunks of 4 (12*4 = 48 classes)

typedef float v2f __attribute__((ext_vector_type(2)));
typedef float v8f __attribute__((ext_vector_type(8)));

// ---------------------------------------------------------------------------
// Kernel 1: exponentiate the transition matrix once (48x48 = 2304 elements).
// ---------------------------------------------------------------------------
__global__ __launch_bounds__(256)
void crf_expT_kernel(const float* __restrict__ trans, float* __restrict__ expT) {
    int i = blockIdx.x * blockDim.x + threadIdx.x;
    if (i < Ccls * Ccls) expT[i] = __expf(trans[i]);
}

// ---------------------------------------------------------------------------
// Kernel 2: scaled forward recursion, one wave (32 lanes) per 16-batch tile.
//   alpha_t = (alpha_{t-1} @ expT) * exp(emit_t), renormalized per row;
//   log_den[b] = sum_t log(rowsum_t).
// 16x48 @ 48x48 per step on V_WMMA_F32_16X16X4_F32 (3 N-tiles x 12 K-chunks).
// Latency engineering (the T-chain is serial, so per-step latency = runtime):
//  - each tile's K accumulation split into 2 independent 6-deep WMMA chains
//    (even/odd chunks) merged with one vector add -> 6 chains in flight;
//  - row sums from D fragments via 4-step __shfl_xor tree per 16-lane half
//    (no serial LDS reduction); LDS always holds *normalized* alpha;
//  - emissions loads issue in one clause ahead of the WMMA chain.
// ---------------------------------------------------------------------------
__global__ __launch_bounds__(32)
void crf_forward_kernel(const float* __restrict__ emissions,
                        const unsigned char* __restrict__ mask,
                        const float* __restrict__ expT,
                        float* __restrict__ log_den) {
    __shared__ float alpha[16 * Ccls];   // normalized alpha tile (rows sum to 1)
    __shared__ unsigned char mk[16];     // per-row mask bit for current t

    const int lane = threadIdx.x;
    const int b0   = blockIdx.x * 16;    // batch rows handled by this wave
    const int hi   = lane >> 4;          // half-wave select (rows 0-7 / 8-15)
    const int lo   = lane & 15;

    // ---- Loop-invariant B fragments: expT tiled for WMMA, kept in VGPRs ----
    // B 4x16 f32 fragment (mirrors documented A layout):
    //   vgpr v, lanes0-15 -> K=v, N=lane ; lanes16-31 -> K=v+2, N=lane-16
    float bfrag[NT][KC][2];
    for (int n = 0; n < NT; ++n)
        for (int k = 0; k < KC; ++k)
            for (int v = 0; v < 2; ++v) {
                int K = 4 * k + v + 2 * hi;
                bfrag[n][k][v] = expT[K * Ccls + n * 16 + lo];
            }

    // ---- t = 0 init in the same fragment layout as the main loop ----
    float vals[NT][8];
    float logZ8[8];                      // log-partition for rows hi*8 .. hi*8+7
    {
        for (int n = 0; n < NT; ++n) {
            int col = n * 16 + lo;
            for (int r = 0; r < 8; ++r) {
                int m = r + 8 * hi;
                vals[n][r] = __expf(
                    emissions[((size_t)(b0 + m) * Tlen) * Ccls + col]);
            }
        }
        float s[8];
        for (int r = 0; r < 8; ++r)
            s[r] = vals[0][r] + vals[1][r] + vals[2][r];
        for (int m = 1; m < 16; m <<= 1)        // xor tree stays inside a half
            for (int r = 0; r < 8; ++r)
                s[r] += __shfl_xor(s[r], m, 32);
        for (int r = 0; r < 8; ++r) {
            float invc = 1.0f / s[r];
            logZ8[r] = __logf(s[r]);
            int m = r + 8 * hi;
            for (int n = 0; n < NT; ++n)
                alpha[m * Ccls + n * 16 + lo] = vals[n][r] * invc;
        }
    }
    __syncthreads();

    // ---- Serial chain over T with WMMA contraction each step ----
    for (int t = 1; t < Tlen; ++t) {
        // A fragments (already normalized). A 16x4 f32:
        //   vgpr v, lanes0-15 -> M=lane, K=v ; lanes16-31 -> M=lane-16, K=v+2
        v2f afrag[KC];
        for (int k = 0; k < KC; ++k) {
            int K = 4 * k + 2 * hi;
            afrag[k][0] = alpha[lo * Ccls + K + 0];
            afrag[k][1] = alpha[lo * Ccls + K + 1];
        }
        // Branch-free mask load: all 32 lanes handle row (lane&15); lanes
        // 16-31 store duplicate bytes (benign) -> no exec-mask juggling.
        mk[lo] = mask[(size_t)(b0 + lo) * Tlen + t];
        __syncthreads();  // old-alpha reads complete before overwrite below

        // Per tile: two independent 6-deep WMMA chains (even/odd K chunks),
        // merged with a vector add -> 6 independent chains across 3 tiles.
        v8f acc[NT];
        for (int n = 0; n < NT; ++n) {
            v8f c0 = {};
            v8f c1 = {};
            for (int k = 0; k < KC; k += 2) {
                v2f be, bo;
                be[0] = bfrag[n][k][0];     be[1] = bfrag[n][k][1];
                bo[0] = bfrag[n][k + 1][0]; bo[1] = bfrag[n][k + 1][1];
                c0 = __builtin_amdgcn_wmma_f32_16x16x4_f32(
                        false, afrag[k],     false, be, (short)0, c0, false, false);
                c1 = __builtin_amdgcn_wmma_f32_16x16x4_f32(
                        false, afrag[k + 1], false, bo, (short)0, c1, false, false);
            }
            acc[n] = c0 + c1;
        }

        // Multiply by exp(emit_t) elementwise (D layout), then reduce rows
        // with the xor-shuffle tree -- no serial LDS reduction.
        for (int n = 0; n < NT; ++n) {
            int col = n * 16 + lo;
            for (int r = 0; r < 8; ++r) {
                int m = r + 8 * hi;
                float e = __expf(
                    emissions[((size_t)(b0 + m) * Tlen + t) * Ccls + col]);
                vals[n][r] = acc[n][r] * e;
            }
        }
        float s[8];
        for (int r = 0; r < 8; ++r)
            s[r] = vals[0][r] + vals[1][r] + vals[2][r];
        for (int m = 1; m < 16; m <<= 1)
            for (int r = 0; r < 8; ++r)
                s[r] += __shfl_xor(s[r], m, 32);

        // Normalize + write back, gated per-row by the mask; accumulate logZ.
        for (int r = 0; r < 8; ++r) {
            int m = r + 8 * hi;
            if (mk[m]) {
                float invc = 1.0f / s[r];
                logZ8[r] += __logf(s[r]);
                for (int n = 0; n < NT; ++n)
                    alpha[m * Ccls + n * 16 + lo] = vals[n][r] * invc;
            }
        }
        // Prefetch next step's emission rows (global_prefetch_b8 path)
        if (t + 1 < Tlen) {
            __builtin_prefetch(
                &emissions[((size_t)(b0 + hi * 8) * Tlen + (t + 1)) * Ccls + lo],
                0, 0);
        }
        __syncthreads();  // new alpha visible for next iteration's A reads
    }

    // logZ8 is identical across the 16 lanes of each half; lanes lo==r write.
    for (int r = 0; r < 8; ++r)
        if (lo == r) log_den[b0 + hi * 8 + r] = logZ8[r];
}

// ---------------------------------------------------------------------------
// Kernel 3: joint (numerator) score, one wave per batch element.
// ---------------------------------------------------------------------------
__global__ __launch_bounds__(32)
void crf_joint_kernel(const float* __restrict__ emissions,
                      const int* __restrict__ tags,
                      const unsigned char* __restrict__ mask,
                      const float* __restrict__ trans,
                      float* __restrict__ log_num) {
    const int b = blockIdx.x;
    const int lane = threadIdx.x;
    float s = 0.0f;
    for (int t = lane; t < Tlen; t += 32) {
        int tg = tags[(size_t)b * Tlen + t];
        unsigned char mt = mask[(size_t)b * Tlen + t];
        if (mt) s += emissions[((size_t)b * Tlen + t) * Ccls + tg];
        if (t >= 1) {
            int tp = tags[(size_t)b * Tlen + t - 1];
            unsigned char mp = mask[(size_t)b * Tlen + t - 1];
            if (mt && mp) s += trans[tp * Ccls + tg];
        }
    }
    for (int off = 16; off > 0; off >>= 1) s += __shfl_down(s, off, 32);
    if (lane == 0) log_num[b] = s;
}

// ---------------------------------------------------------------------------
// Kernel 4: deterministic final reduction -> mean(log_den - log_num)
// ---------------------------------------------------------------------------
__global__ __launch_bounds__(256)
void crf_reduce_kernel(const float* __restrict__ log_den,
                       const float* __restrict__ log_num,
                       float* __restrict__ out) {
    __shared__ float sm[256];
    const int tid = threadIdx.x;
    float s = 0.0f;
    for (int i = tid; i < Bsz; i += 256) s += log_den[i] - log_num[i];
    sm[tid] = s;
    __syncthreads();
    for (int off = 128; off > 0; off >>= 1) {
        if (tid < off) sm[tid] += sm[tid + off];
        __syncthreads();
    }
    if (tid == 0) out[0] = sm[0] / (float)Bsz;
}

// ---------------------------------------------------------------------------
extern "C" void kernel_launch(void* const* d_in, const int* in_sizes, int n_in,
                              void* d_out, int out_size, void* d_ws, size_t ws_size,
                              hipStream_t stream) {
    const float*         emissions   = (const float*)d_in[0];
    const int*           tags        = (const int*)d_in[1];
    const unsigned char* mask        = (const unsigned char*)d_in[2];
    const float*         transitions = (const float*)d_in[3];

    float* expT    = (float*)d_ws;             // 48*48 floats
    float* log_den = expT + Ccls * Ccls;       // 512 floats
    float* log_num = log_den + Bsz;            // 512 floats

    crf_expT_kernel<<<(Ccls * Ccls + 255) / 256, 256, 0, stream>>>(transitions, expT);
    crf_forward_kernel<<<Bsz / 16, 32, 0, stream>>>(emissions, mask, expT, log_den);
    crf_joint_kernel<<<Bsz, 32, 0, stream>>>(emissions, tags, mask, transitions, log_num);
    crf_reduce_kernel<<<1, 256, 0, stream>>>(log_den, log_num, (float*)d_out);
}